// RGCNGRU_18511309046057
// MI455X (gfx1250) — compile-verified
//
#include <hip/hip_runtime.h>
#include <hip/hip_bf16.h>

// GConvGRU(K=1), H0=0  ==>  pure dense per-node math:
//   Z  = sigmoid(x @ Wxz + bxz + bhz)
//   Ht = tanh  (x @ Wxh + bxh + bhh)
//   out= relu((1-Z)*Ht) @ Wlin + blin
// edge_index / edge_weight / Wxr / Whr / Whz / Whh are mathematically unused.
// Memory-bound: 64MB x reads + 4MB out writes @ 23.3 TB/s  (~3 us floor).
// GEMMs via CDNA5 fp32 WMMA: 16x V_WMMA_F32_16X16X4_F32 per wave-tile.
// Activations are branch-free v_exp_f32 + v_rcp_f32 (no libm tanh/div expansion).

typedef float v2f __attribute__((ext_vector_type(2)));
typedef float v8f __attribute__((ext_vector_type(8)));

#define LOG2E 1.4426950408889634f

__device__ __forceinline__ float fast_sigmoid(float v) {
    // 1 / (1 + e^-v) ; exp2 overflow/underflow saturates correctly to 0/1
    return __builtin_amdgcn_rcpf(1.0f + __builtin_amdgcn_exp2f(-v * LOG2E));
}

__device__ __forceinline__ float fast_tanh(float v) {
    // 1 - 2/(e^{2v}+1) ; saturates to +/-1 at the exp2 inf/0 ends
    float e = __builtin_amdgcn_exp2f(v * (2.0f * LOG2E));
    return fmaf(-2.0f, __builtin_amdgcn_rcpf(e + 1.0f), 1.0f);
}

__global__ __launch_bounds__(256) void gconvgru_wmma_kernel(
    const float* __restrict__ x,
    const float* __restrict__ Wxz, const float* __restrict__ bxz, const float* __restrict__ bhz,
    const float* __restrict__ Wxh, const float* __restrict__ bxh, const float* __restrict__ bhh,
    const float* __restrict__ Wlin, const float* __restrict__ blin,
    float* __restrict__ out, int N)
{
    const int lane = threadIdx.x & 31;
    const int wave = (int)((blockIdx.x * blockDim.x + threadIdx.x) >> 5);
    const int base = wave << 4;              // first node of this wave's 16-node tile
    if (base >= N) return;                   // wave-uniform guard: EXEC stays all-1s

    const int hi = lane >> 4;                // 0: lanes 0-15, 1: lanes 16-31
    const int m  = lane & 15;

    // ---------------- A fragments: x tile [16 nodes x 16 feats], four 16x4 chunks
    // ISA layout (32-bit A 16x4): lanes 0-15 hold K=4k,4k+1 ; lanes 16-31 hold K=4k+2,4k+3
    int row = base + m;
    if (row >= N) row = N - 1;               // clamp tail (keeps EXEC full for WMMA)
    const float* xr = x + (size_t)row * 16;
    v2f a[4];
#pragma unroll
    for (int kc = 0; kc < 4; ++kc) {
        const float* p = xr + 4 * kc + 2 * hi;
        a[kc].x = p[0];
        a[kc].y = p[1];
    }

    // ---------------- B fragments: W[16 x 32] row-major, two 16-col tiles per gate
    // ISA layout (32-bit B 4x16): lane = column N; lanes 0-15 hold K=4k,4k+1 ;
    // lanes 16-31 hold K=4k+2,4k+3 (VGPR0 = even K of the pair, VGPR1 = odd).
    v2f bz0[4], bz1[4], bh0[4], bh1[4];
#pragma unroll
    for (int kc = 0; kc < 4; ++kc) {
        const int k0 = 4 * kc + 2 * hi;
        const int n0 = m;
        const int n1 = m + 16;
        bz0[kc].x = Wxz[(k0    ) * 32 + n0]; bz0[kc].y = Wxz[(k0 + 1) * 32 + n0];
        bz1[kc].x = Wxz[(k0    ) * 32 + n1]; bz1[kc].y = Wxz[(k0 + 1) * 32 + n1];
        bh0[kc].x = Wxh[(k0    ) * 32 + n0]; bh0[kc].y = Wxh[(k0 + 1) * 32 + n0];
        bh1[kc].x = Wxh[(k0    ) * 32 + n1]; bh1[kc].y = Wxh[(k0 + 1) * 32 + n1];
    }

    // ---------------- 16x WMMA f32 16x16x4: two gates x two n-tiles x four k-chunks
    v8f az0 = {}, az1 = {}, ah0 = {}, ah1 = {};
#pragma unroll
    for (int kc = 0; kc < 4; ++kc) {
        az0 = __builtin_amdgcn_wmma_f32_16x16x4_f32(false, a[kc], false, bz0[kc], (short)0, az0, false, false);
        az1 = __builtin_amdgcn_wmma_f32_16x16x4_f32(false, a[kc], false, bz1[kc], (short)0, az1, false, false);
        ah0 = __builtin_amdgcn_wmma_f32_16x16x4_f32(false, a[kc], false, bh0[kc], (short)0, ah0, false, false);
        ah1 = __builtin_amdgcn_wmma_f32_16x16x4_f32(false, a[kc], false, bh1[kc], (short)0, ah1, false, false);
    }

    // ---------------- epilogue: biases, activations, GRU combine, relu, dot(Wlin)
    // C/D layout: VGPR r -> row M=r (lanes 0-15) / M=r+8 (lanes 16-31), col = lane&15 (+16*tile)
    const float bz_n0 = bxz[m]      + bhz[m];
    const float bz_n1 = bxz[m + 16] + bhz[m + 16];
    const float bh_n0 = bxh[m]      + bhh[m];
    const float bh_n1 = bxh[m + 16] + bhh[m + 16];
    const float wl0   = Wlin[m];
    const float wl1   = Wlin[m + 16];
    const float bl    = blin[0];

#pragma unroll
    for (int r = 0; r < 8; ++r) {
        float z0 = fast_sigmoid(az0[r] + bz_n0);
        float z1 = fast_sigmoid(az1[r] + bz_n1);
        float t0 = fast_tanh(ah0[r] + bh_n0);
        float t1 = fast_tanh(ah1[r] + bh_n1);
        float h0 = fmaxf((1.0f - z0) * t0, 0.0f);
        float h1 = fmaxf((1.0f - z1) * t1, 0.0f);
        float p  = fmaf(h0, wl0, h1 * wl1);
        // reduce across the 16 lanes of this half-wave (one output row each)
#pragma unroll
        for (int off = 8; off >= 1; off >>= 1)
            p += __shfl_xor(p, off, 16);
        const int node = base + r + 8 * hi;      // lanes 0-15 -> row r, lanes 16-31 -> row r+8
        if (m == 0 && node < N)
            out[node] = p + bl;
    }
}

extern "C" void kernel_launch(void* const* d_in, const int* in_sizes, int n_in,
                              void* d_out, int out_size, void* d_ws, size_t ws_size,
                              hipStream_t stream) {
    // setup_inputs() order:
    // 0:x 1:edge_index 2:edge_weight 3:Wxz 4:bxz 5:Whz 6:bhz 7:Wxr 8:bxr 9:Whr
    // 10:bhr 11:Wxh 12:bxh 13:Whh 14:bhh 15:Wlin 16:blin
    const float* x    = (const float*)d_in[0];
    const float* Wxz  = (const float*)d_in[3];
    const float* bxz  = (const float*)d_in[4];
    const float* bhz  = (const float*)d_in[6];
    const float* Wxh  = (const float*)d_in[11];
    const float* bxh  = (const float*)d_in[12];
    const float* bhh  = (const float*)d_in[14];
    const float* Wlin = (const float*)d_in[15];
    const float* blin = (const float*)d_in[16];
    float* out = (float*)d_out;

    const int N = in_sizes[0] / 16;
    const int waves = (N + 15) / 16;             // one wave32 per 16-node tile
    const int threads = 256;                      // 8 waves per workgroup
    const int wavesPerBlock = threads / 32;
    const int blocks = (waves + wavesPerBlock - 1) / wavesPerBlock;

    gconvgru_wmma_kernel<<<blocks, threads, 0, stream>>>(
        x, Wxz, bxz, bhz, Wxh, bxh, bhh, Wlin, blin, out, N);
}